// SequentialLoss_6390911337254
// MI455X (gfx1250) — compile-verified
//
#include <hip/hip_runtime.h>
#include <stdint.h>

// Problem constants (match reference)
#define BB      128
#define SS      4096
#define CC      64          // classes per row
#define NROWS   (BB * SS)   // 524288 rows, 64 f32 each

// Streaming configuration
#define THREADS          128                      // 4 waves of 32
#define TILE_ROWS        128                      // one row per thread per tile
#define TILE_F4          (TILE_ROWS * (CC / 4))   // 2048 float4 = 32 KB
#define F4_PER_THREAD    (TILE_F4 / THREADS)      // 16 async b128 per thread per tile
#define NBLOCKS          1024
#define TILES_PER_BLOCK  (NROWS / TILE_ROWS / NBLOCKS)  // 4

typedef __attribute__((ext_vector_type(2))) float v2f;
typedef __attribute__((ext_vector_type(8))) float v8f;

// Low 32 bits of a flat shared-aperture address == LDS byte offset.
__device__ __forceinline__ uint32_t lds_off_of(const void* p) {
    return (uint32_t)(uintptr_t)p;
}

// CDNA5 async tensor/data path: global -> LDS, tracked by ASYNCcnt.
__device__ __forceinline__ void async_copy_b128(uint32_t lds_byte_off, const float4* g) {
    asm volatile("global_load_async_to_lds_b128 %0, %1, off"
                 :: "v"(lds_byte_off), "v"((uint64_t)(uintptr_t)g)
                 : "memory");
}

__device__ __forceinline__ void stage_tile(const float4* gbase, float* sbuf, int t) {
    uint32_t base = lds_off_of(sbuf);
#pragma unroll
    for (int j = 0; j < F4_PER_THREAD; ++j) {
        int f = j * THREADS + t;                 // coalesced across the block
        async_copy_b128(base + (uint32_t)f * 16u, gbase + f);
    }
}

__global__ __launch_bounds__(THREADS)
void seq_loss_main(const float* __restrict__ x,
                   const int*   __restrict__ target,
                   const int*   __restrict__ lengths,
                   float*       __restrict__ partials) {
    __shared__ float sbuf[2][TILE_ROWS * CC];    // 2 x 32 KB double buffer
    __shared__ float red[THREADS];

    const int t    = threadIdx.x;
    const int lane = t & 31;
    const int tile0 = blockIdx.x * TILES_PER_BLOCK;
    const float4* xg = (const float4*)x;

    // Prologue: prefetch tiles 0 and 1 (16 + 16 async ops outstanding per wave)
    stage_tile(xg + (size_t)tile0 * TILE_F4,       sbuf[0], t);
    stage_tile(xg + (size_t)(tile0 + 1) * TILE_F4, sbuf[1], t);

    float acc = 0.0f;

#pragma unroll
    for (int i = 0; i < TILES_PER_BLOCK; ++i) {
        // Drain this tile's async loads, keep the next tile's 16 in flight.
        if (i + 1 < TILES_PER_BLOCK) {
            asm volatile("s_wait_asynccnt 0x10" ::: "memory");
        } else {
            asm volatile("s_wait_asynccnt 0x0" ::: "memory");
        }
        __syncthreads();

        const float* row = &sbuf[i & 1][t * CC];
        const int gr   = (tile0 + i) * TILE_ROWS + t;   // global row
        const int b    = gr >> 12;                      // / S (4096)
        const int sidx = gr & (SS - 1);                 // % S
        const int tgt  = target[b];
        const int tgt4 = tgt >> 2;
        const int tr   = tgt & 3;

        // Pass 1: row max. Rotation (k+lane)&15 -> conflict-free ds_load_b128.
        float m = -INFINITY;
#pragma unroll
        for (int k = 0; k < CC / 4; ++k) {
            const int c4 = (k + lane) & 15;
            const float4 v = *(const float4*)(row + c4 * 4);
            m = fmaxf(m, fmaxf(fmaxf(v.x, v.y), fmaxf(v.z, v.w)));
        }

        // Pass 2: sum of exp + free capture of x[target] during the rotation.
        float ssum = 0.0f;
        float xt   = 0.0f;
#pragma unroll
        for (int k = 0; k < CC / 4; ++k) {
            const int c4 = (k + lane) & 15;
            const float4 v = *(const float4*)(row + c4 * 4);
            ssum += __expf(v.x - m);
            ssum += __expf(v.y - m);
            ssum += __expf(v.z - m);
            ssum += __expf(v.w - m);
            if (c4 == tgt4) {
                const float lo = (tr & 1) ? v.y : v.x;
                const float hi = (tr & 1) ? v.w : v.z;
                xt = (tr & 2) ? hi : lo;
            }
        }

        const float nll = (m + __logf(ssum)) - xt;      // logsumexp - x[tgt]
        const int   Li  = lengths[b];
        const float Lf  = (float)Li;
        const float w   = (sidx < Li) ? ((float)sidx / fmaxf(Lf, 1.0f)) : 0.0f;
        acc += w * nll;

        __syncthreads();                                // everyone done reading buffer
        if (i + 2 < TILES_PER_BLOCK)                    // refill it
            stage_tile(xg + (size_t)(tile0 + i + 2) * TILE_F4, sbuf[i & 1], t);
    }

    // Deterministic block reduction.
    red[t] = acc;
    __syncthreads();
#pragma unroll
    for (int s2 = THREADS >> 1; s2 > 0; s2 >>= 1) {
        if (t < s2) red[t] += red[t + s2];
        __syncthreads();
    }
    if (t == 0) partials[blockIdx.x] = red[0];
}

// Final reduction: 1024 partials -> scalar. One full wave32; the cross-lane
// fold uses v_wmma_f32_16x16x4_f32 with an all-ones B (row-sum reduction).
__global__ __launch_bounds__(32)
void seq_loss_final(const float* __restrict__ partials, float* __restrict__ out) {
    const int lane = threadIdx.x;                 // 0..31, EXEC all ones
    float s = 0.0f;
#pragma unroll
    for (int k = 0; k < NBLOCKS / 32; ++k)
        s += partials[k * 32 + lane];

    // A (16x4 f32): VGPR0 = lane value (K=0 for lanes 0-15, K=2 for 16-31),
    // VGPR1 = 0. B = ones(4x16)  =>  D[m][n] = s[m] + s[m+16] for every n.
    v2f a;  a[0] = s;    a[1] = 0.0f;
    v2f bo; bo[0] = 1.0f; bo[1] = 1.0f;
    v8f c = {};
    c = __builtin_amdgcn_wmma_f32_16x16x4_f32(false, a, false, bo,
                                              (short)0, c, false, false);
    // lane n<16 holds D[0..7][n]; lane n>=16 holds D[8..15][n-16].
    float r = c[0] + c[1] + c[2] + c[3] + c[4] + c[5] + c[6] + c[7];
    r += __shfl_xor(r, 16, 32);                   // lanes 0 and 16 combine halves
    if (lane == 0) out[0] = r / (float)BB;        // mean over batch
}

extern "C" void kernel_launch(void* const* d_in, const int* in_sizes, int n_in,
                              void* d_out, int out_size, void* d_ws, size_t ws_size,
                              hipStream_t stream) {
    const float* x       = (const float*)d_in[0];   // [B,S,C] logits
    const int*   target  = (const int*)d_in[1];     // [B]
    const int*   lengths = (const int*)d_in[2];     // [B]
    float*       partials = (float*)d_ws;           // NBLOCKS floats of scratch
    float*       outp     = (float*)d_out;          // scalar f32

    seq_loss_main<<<dim3(NBLOCKS), dim3(THREADS), 0, stream>>>(x, target, lengths, partials);
    seq_loss_final<<<dim3(1), dim3(32), 0, stream>>>(partials, outp);
}